// SAGPoolingLayer_57655640982223
// MI455X (gfx1250) — compile-verified
//
#include <hip/hip_runtime.h>
#include <hip/hip_bf16.h>
#include <math.h>

typedef float v2f __attribute__((ext_vector_type(2)));
typedef float v8f __attribute__((ext_vector_type(8)));

#define D 256               // feature dim
#define NPG 512             // nodes per graph

// ---------------------------------------------------------------------------
// Kernel 1: per-node dual dot products.  r_i = x_i . w_rel, s_i = x_i . w_root
// One wave32 per node: lane loads 8 contiguous floats (coalesced b128 loads),
// wave-level shfl reduction.
// ---------------------------------------------------------------------------
__global__ void node_dots(const float* __restrict__ x,
                          const float* __restrict__ w_rel,
                          const float* __restrict__ w_root,
                          float* __restrict__ r, float* __restrict__ s, int N) {
    const int lane = threadIdx.x & 31;
    const int wave = threadIdx.x >> 5;
    const int node = blockIdx.x * (blockDim.x >> 5) + wave;
    if (node >= N) return;
    const float* xr = x + (size_t)node * D;
    const int j0 = lane * 8;                 // 32 lanes * 8 = 256 = D
    float ar = 0.f, as = 0.f;
#pragma unroll
    for (int t = 0; t < 8; ++t) {
        const float xv = xr[j0 + t];
        ar += xv * w_rel[j0 + t];
        as += xv * w_root[j0 + t];
    }
#pragma unroll
    for (int off = 16; off > 0; off >>= 1) {
        ar += __shfl_down(ar, off);
        as += __shfl_down(as, off);
    }
    if (lane == 0) { r[node] = ar; s[node] = as; }
}

// ---------------------------------------------------------------------------
// Kernel 2: one workgroup per graph.
//  (a) scalar scatter-add over this graph's edges into LDS (ds_add_f32)
//  (b) score = tanh(acc + s + b_rel)
//  (c) exact top-k via rank counting (matches jax.lax.top_k tie-breaking)
//  (d) pooled[c] = (1/k) * sum_{kept i} score[i] * x[i][c]
// ---------------------------------------------------------------------------
__global__ void graph_score_pool(const float* __restrict__ x,
                                 const float* __restrict__ r,
                                 const float* __restrict__ s,
                                 const int* __restrict__ esrc,
                                 const int* __restrict__ edst,
                                 const float* __restrict__ b_rel,
                                 float* __restrict__ pooled,
                                 int Eg, int k) {
    __shared__ float acc[NPG];
    __shared__ float score[NPG];
    __shared__ int   keep[NPG];
    __shared__ float partial[2 * D];

    const int b     = blockIdx.x;
    const int tid   = threadIdx.x;           // blockDim.x == 512 == NPG
    const int nbase = b * NPG;

    acc[tid] = 0.f;
    __syncthreads();

    const long ebase = (long)b * Eg;         // edges of graph b are contiguous
    for (int e = tid; e < Eg; e += blockDim.x) {
        const int sg = esrc[ebase + e];          // global src node
        const int dg = edst[ebase + e] - nbase;  // local dst node
        atomicAdd(&acc[dg], r[sg]);              // ds_add_f32
    }
    __syncthreads();

    const float sc = tanhf(acc[tid] + s[nbase + tid] + b_rel[0]);
    score[tid] = sc;
    __syncthreads();

    // rank_i = #{score_j > score_i} + #{j<i : score_j == score_i}; keep if < k
    int cnt = 0;
    for (int j = 0; j < NPG; ++j) {
        const float v = score[j];
        cnt += (v > sc) || (v == sc && j < tid);
    }
    keep[tid] = (cnt < k) ? 1 : 0;
    __syncthreads();

    // weighted pooling: 256 threads per row-half, thread owns one column
    const int c = tid & (D - 1);
    const int h = tid >> 8;                  // row-half: 0 or 1
    const int i0 = h * (NPG / 2);
    float ap = 0.f;
    for (int i = i0; i < i0 + NPG / 2; ++i) {
        if (keep[i])                         // uniform across the half -> no divergence
            ap += score[i] * x[(size_t)(nbase + i) * D + c];
    }
    partial[h * D + c] = ap;
    __syncthreads();
    if (tid < D)
        pooled[(size_t)b * D + tid] = (partial[tid] + partial[D + tid]) / (float)k;
}

// ---------------------------------------------------------------------------
// Kernel 3: out[M,N] = pooled[M,K] @ w_proj[K,N] + bias, f32 WMMA.
// One wave per 16x16 output tile, K-loop in steps of 4 using
// V_WMMA_F32_16X16X4_F32.  A 16x4 layout: lanes 0-15 hold K={0,1} in
// VGPR{0,1}, lanes 16-31 hold K={2,3}; B 4x16 mirrors with lanes = columns.
// ---------------------------------------------------------------------------
__global__ void proj_wmma(const float* __restrict__ A,
                          const float* __restrict__ Bm,
                          const float* __restrict__ bias,
                          float* __restrict__ out,
                          int M, int K, int Nn) {
    const int lane  = threadIdx.x & 31;
    const int half  = lane >> 4;             // 0: lanes 0-15, 1: lanes 16-31
    const int l16   = lane & 15;
    const int tileM = blockIdx.x;
    const int tileN = blockIdx.y;

    const int rowA = tileM * 16 + l16;       // A row owned by this lane
    const int colB = tileN * 16 + l16;       // B col owned by this lane

    v8f c = {};
    for (int k0 = 0; k0 < K; k0 += 4) {
        const int ka = k0 + half * 2;
        v2f a, bf;
        a.x  = A[(size_t)rowA * K + ka];
        a.y  = A[(size_t)rowA * K + ka + 1];
        bf.x = Bm[(size_t)ka * Nn + colB];
        bf.y = Bm[(size_t)(ka + 1) * Nn + colB];
        // (neg_a, A, neg_b, B, c_mod, C, reuse_a, reuse_b)
        c = __builtin_amdgcn_wmma_f32_16x16x4_f32(false, a, false, bf,
                                                  (short)0, c, false, false);
    }

    const float bv = bias[colB];
#pragma unroll
    for (int v = 0; v < 8; ++v) {
        const int row = tileM * 16 + v + half * 8;   // C/D: VGPR v -> M=v / v+8
        out[(size_t)row * Nn + colB] = c[v] + bv;
    }
}

// ---------------------------------------------------------------------------
// Host-side launcher
// ---------------------------------------------------------------------------
extern "C" void kernel_launch(void* const* d_in, const int* in_sizes, int n_in,
                              void* d_out, int out_size, void* d_ws, size_t ws_size,
                              hipStream_t stream) {
    const float* x      = (const float*)d_in[0];
    const int*   eidx   = (const int*)d_in[1];   // [2, E] flat
    // d_in[2] = batch (unused: graphs are contiguous blocks)
    // d_in[3] = num_graphs (derived from sizes instead)
    const float* w_rel  = (const float*)d_in[4];
    const float* b_rel  = (const float*)d_in[5];
    const float* w_root = (const float*)d_in[6];
    const float* w_proj = (const float*)d_in[7];
    const float* b_proj = (const float*)d_in[8];
    float*       out    = (float*)d_out;

    const int d    = D;
    const int N    = in_sizes[0] / d;        // total nodes (65536)
    const int Bg   = N / NPG;                // graphs (128)
    const int E    = in_sizes[1] / 2;        // edges (524288)
    const int Eg   = E / Bg;                 // edges per graph (4096)
    const int k    = (NPG + 1) / 2;          // ceil(0.5 * n) = 256
    const int outd = in_sizes[8];            // out_dim = 256

    float* r      = (float*)d_ws;            // [N]
    float* s      = r + N;                   // [N]
    float* pooled = s + N;                   // [Bg, d]

    // K1: 8 waves (8 nodes) per 256-thread block
    node_dots<<<dim3((N + 7) / 8), dim3(256), 0, stream>>>(x, w_rel, w_root, r, s, N);

    // K2: one block per graph
    graph_score_pool<<<dim3(Bg), dim3(NPG), 0, stream>>>(
        x, r, s, eidx, eidx + E, b_rel, pooled, Eg, k);

    // K3: one wave per 16x16 tile of the [Bg, outd] output
    proj_wmma<<<dim3(Bg / 16, outd / 16), dim3(32), 0, stream>>>(
        pooled, w_proj, b_proj, out, Bg, d, outd);
}